// Decoding_model_23570780521049
// MI455X (gfx1250) — compile-verified
//
#include <hip/hip_runtime.h>
#include <hip/hip_bf16.h>
#include <math.h>

// NMS LDPC decoder, (63,127) code, row weight 8, batch 4096, 5 iterations.
// Bandwidth-bound (~18.6 MB total traffic -> ~0.8us @ 23.3 TB/s on MI455X).
// Strategy: stage each 8-row batch tile into LDS once via the CDNA5 async
// path (global_load_async_to_lds_b32 + s_wait_asynccnt), run all 5 decoder
// iterations out of LDS, stream only mandatory outputs coalesced.
// No WMMA on purpose: densifying H (63x127, 8/127 sparse) would add
// ~130 MB/iter of intermediate traffic on a purely bandwidth-bound problem.

#define NCODE  127
#define MCHK   63
#define RW     8
#define NEDGE  (MCHK * RW)        // 504
#define NITER  5
#define BATCHN 4096
#define ROWS   8                  // batch rows per block
#define TPB    512                // 16 waves (wave32)
#define TILE   (ROWS * NCODE)     // 1016 elements (<= 2*TPB)
#define STRIDE 129                // odd stride -> conflict-free LDS banks
#define MAXDEG 32
#define BIGV   1.0e10f

#ifndef __has_builtin
#define __has_builtin(x) 0
#endif

#if __has_builtin(__builtin_amdgcn_global_load_async_to_lds_b32)
#define HAVE_ASYNC 1
#else
#define HAVE_ASYNC 0
#endif

typedef __attribute__((address_space(1))) int  gint_t;
typedef __attribute__((address_space(3))) int  lint_t;

__global__ __launch_bounds__(TPB) void nms_decode_kernel(
    const float* __restrict__ soft, const int* __restrict__ labels,
    const int* __restrict__ H, const float* __restrict__ normz,
    float* __restrict__ out)
{
    __shared__ float cur[ROWS][STRIDE];        // LDS-resident state
    __shared__ float edgebuf[ROWS][NEDGE];     // per-edge messages
    __shared__ int   sidx[NEDGE];              // check -> 8 column indices
    __shared__ int   col_cnt[NCODE];
    __shared__ int   col_list[NCODE][MAXDEG];  // column -> incident edges (ordered)

    const int    tid     = threadIdx.x;
    const int    baseRow = blockIdx.x * ROWS;
    const size_t gbase   = (size_t)baseRow * NCODE;

    // Each thread owns at most 2 tile elements; hoist (r,j) decomposition so
    // the div/mod-by-127 magic sequences never appear inside the hot loops.
    const int  k0 = tid;
    const int  k1 = tid + TPB;
    const int  r0 = k0 / NCODE, j0 = k0 - r0 * NCODE;
    const bool e1 = (k1 < TILE);
    const int  r1 = e1 ? (k1 / NCODE) : 0;
    const int  j1 = e1 ? (k1 - r1 * NCODE) : 0;

    // ---- Stage soft_input tile into LDS (async DMA on CDNA5) ----
#if HAVE_ASYNC
    __builtin_amdgcn_global_load_async_to_lds_b32(
        (gint_t*)(soft + gbase + k0), (lint_t*)&cur[r0][j0], 0, 0);
    if (e1)
        __builtin_amdgcn_global_load_async_to_lds_b32(
            (gint_t*)(soft + gbase + k1), (lint_t*)&cur[r1][j1], 0, 0);
#else
    cur[r0][j0] = soft[gbase + k0];
    if (e1) cur[r1][j1] = soft[gbase + k1];
#endif

    // ---- Build check adjacency from H (63x127 row-major; H stays hot in L2) ----
    if (tid < MCHK) {
        int cnt = 0;
        for (int j = 0; j < NCODE; ++j)
            if (H[tid * NCODE + j] != 0 && cnt < RW) sidx[tid * RW + cnt++] = j;
    }

#if HAVE_ASYNC
#if __has_builtin(__builtin_amdgcn_s_wait_asynccnt)
    __builtin_amdgcn_s_wait_asynccnt(0);
#else
    asm volatile("s_wait_asynccnt 0" ::: "memory");
#endif
#endif
    __syncthreads();

    // ---- Column adjacency, ascending edge order => deterministic sums ----
    if (tid < NCODE) {
        int cnt = 0;
        for (int e = 0; e < NEDGE; ++e)
            if (sidx[e] == tid && cnt < MAXDEG) col_list[tid][cnt++] = e;
        col_cnt[tid] = cnt;
    }

    // ---- outs[0] = soft_input ; labels passthrough (bitwise) ----
    out[gbase + k0] = cur[r0][j0];
    if (e1) out[gbase + k1] = cur[r1][j1];
    {
        int* lab_out = (int*)(out + (size_t)(NITER + 1) * BATCHN * NCODE);
        lab_out[gbase + k0] = labels[gbase + k0];
        if (e1) lab_out[gbase + k1] = labels[gbase + k1];
    }

    // norm = softplus(normalizor) in JAX's stable form
    const float x    = normz[0];
    const float norm = fmaxf(x, 0.0f) + log1pf(expf(-fabsf(x)));

    const int  r      = tid >> 6;          // 0..7  : batch row within tile
    const int  ci     = tid & 63;          // 0..63 : check index (63 active)
    const bool is_chk = (ci < MCHK);
    __syncthreads();

    // Hoist the 8 column indices of this thread's check (loop-invariant).
    int id[RW];
#pragma unroll
    for (int k = 0; k < RW; ++k)
        id[k] = is_chk ? sidx[ci * RW + k] : 0;

    for (int it = 1; it <= NITER; ++it) {
        // ---- check-node update: per-edge messages into edgebuf ----
        if (is_chk) {
            float v[RW];
            float P = 1.0f, m1 = BIGV, m2 = BIGV;
#pragma unroll
            for (int k = 0; k < RW; ++k) {
                float val = cur[r][id[k]];
                v[k] = val;
                float s = (val > 0.0f) ? 1.0f : ((val < 0.0f) ? -1.0f : 0.0f);
                P *= s;                                  // sign(0)=0 zeroes the check
                float av = fabsf(val);
                float pr = (av == 0.0f) ? BIGV : av;     // reference's BIG masking
                if (pr < m1) { m2 = m1; m1 = pr; }
                else if (pr < m2) { m2 = pr; }           // duplicate-min => second==min
            }
#pragma unroll
            for (int k = 0; k < RW; ++k) {
                float av  = fabsf(v[k]);
                float s   = (v[k] > 0.0f) ? 1.0f : ((v[k] < 0.0f) ? -1.0f : 0.0f);
                float upd = (av == m1) ? m2 : m1;        // exclude-self via tie rule
                edgebuf[r][ci * RW + k] = (norm * upd) * (P * s);
            }
        }
        __syncthreads();

        // ---- variable-node update + coalesced slab store ----
        float* slab = out + (size_t)it * BATCHN * NCODE;
        {
            float acc = 0.0f;
            const int cnt = col_cnt[j0];
            for (int t = 0; t < cnt; ++t) acc += edgebuf[r0][col_list[j0][t]];
            const float nv = cur[r0][j0] + acc;
            cur[r0][j0] = nv;
            slab[gbase + k0] = nv;
        }
        if (e1) {
            float acc = 0.0f;
            const int cnt = col_cnt[j1];
            for (int t = 0; t < cnt; ++t) acc += edgebuf[r1][col_list[j1][t]];
            const float nv = cur[r1][j1] + acc;
            cur[r1][j1] = nv;
            slab[gbase + k1] = nv;
        }
        __syncthreads();
    }
}

extern "C" void kernel_launch(void* const* d_in, const int* in_sizes, int n_in,
                              void* d_out, int out_size, void* d_ws, size_t ws_size,
                              hipStream_t stream) {
    (void)in_sizes; (void)n_in; (void)out_size; (void)d_ws; (void)ws_size;
    const float* soft   = (const float*)d_in[0];   // (4096,127) f32
    const int*   labels = (const int*)d_in[1];     // (4096,127) i32
    const int*   H      = (const int*)d_in[2];     // (63,127)   i32
    const float* normz  = (const float*)d_in[3];   // (1,)       f32
    float*       out    = (float*)d_out;           // 6*4096*127 f32 + 4096*127 i32 bits

    nms_decode_kernel<<<BATCHN / ROWS, TPB, 0, stream>>>(soft, labels, H, normz, out);
}